// ClassChannelAttention_89970974916930
// MI455X (gfx1250) — compile-verified
//
#include <hip/hip_runtime.h>
#include <hip/hip_bf16.h>
#include <math.h>

typedef __attribute__((ext_vector_type(2))) float v2f;
typedef __attribute__((ext_vector_type(4))) float v4f;
typedef __attribute__((ext_vector_type(8))) float v8f;

#define K_CLASSES 150
#define C_CHANNELS 768

// ---------------------------------------------------------------------------
// Kernel A: per-row softmax stats. One wave32 per class row.
//   m[k]    = max_c A[k, c]
//   invd[k] = 1 / sum_c exp(A[k, c] - m[k])
// ---------------------------------------------------------------------------
__global__ void cca_rowstats(const float* __restrict__ attn,
                             float* __restrict__ m,
                             float* __restrict__ invd) {
    const int k = blockIdx.x;          // 0..149
    const int lane = threadIdx.x;      // 0..31 (wave32)
    const float* row = attn + k * C_CHANNELS;

    float mx = -INFINITY;
    for (int c = lane; c < C_CHANNELS; c += 32)
        mx = fmaxf(mx, row[c]);
    for (int off = 16; off > 0; off >>= 1)
        mx = fmaxf(mx, __shfl_xor(mx, off, 32));

    float s = 0.0f;
    for (int c = lane; c < C_CHANNELS; c += 32)
        s += __expf(row[c] - mx);
    for (int off = 16; off > 0; off >>= 1)
        s += __shfl_xor(s, off, 32);

    if (lane == 0) {
        m[k]    = mx;
        invd[k] = 1.0f / s;
    }
}

// ---------------------------------------------------------------------------
// Kernel B: scale[c] = sum_k softmax(A)[k, c]  via V_WMMA_F32_16X16X4_F32.
// ones(16x4) @ P(4x16) accumulated over 38 K-chunks. A = all-ones makes every
// output row equal to the column sum, so the result is invariant to the exact
// K-slot assignment inside the B operand VGPRs; only lane->column matters.
// One wave per 16-column tile; 48 tiles cover C=768.
// ---------------------------------------------------------------------------
__global__ void cca_scale_wmma(const float* __restrict__ attn,
                               const float* __restrict__ m,
                               const float* __restrict__ invd,
                               float* __restrict__ scale) {
    const int lane = threadIdx.x;              // 0..31
    const int tile = blockIdx.x;               // 0..47
    const int col  = tile * 16 + (lane & 15);  // output channel for this lane
    const int kgrp = (lane >> 4) * 2;          // lanes 0-15 -> K pair {0,1}; 16-31 -> {2,3}

    const v2f a_ones = {1.0f, 1.0f};
    v8f acc = {};

    for (int kc = 0; kc < 152; kc += 4) {      // 38 chunks of K=4 (pad 150 -> 152)
        const int k0 = kc + kgrp;
        const int k1 = k0 + 1;
        // Clamp + mask instead of branching so EXEC stays all-ones at the WMMA.
        const int k0c = k0 < K_CLASSES ? k0 : K_CLASSES - 1;
        const int k1c = k1 < K_CLASSES ? k1 : K_CLASSES - 1;
        float p0 = __expf(attn[k0c * C_CHANNELS + col] - m[k0c]) * invd[k0c];
        float p1 = __expf(attn[k1c * C_CHANNELS + col] - m[k1c]) * invd[k1c];
        p0 = (k0 < K_CLASSES) ? p0 : 0.0f;
        p1 = (k1 < K_CLASSES) ? p1 : 0.0f;
        const v2f b = {p0, p1};
        // D = ones(16x4) * B(4x16) + C  -> every row of D is the column sum.
        acc = __builtin_amdgcn_wmma_f32_16x16x4_f32(
            /*neg_a=*/false, a_ones, /*neg_b=*/false, b,
            /*c_mod=*/(short)0, acc, /*reuse_a=*/false, /*reuse_b=*/false);
    }

    // C/D layout: VGPR0, lanes 0-15 hold M=0, N=lane.
    if (lane < 16)
        scale[col] = acc[0];
}

// ---------------------------------------------------------------------------
// Kernel C: out = x * scale[c]. Pure streaming, 16B per lane per direction.
// Channel index is uniform per block -> scale[] read scalarizes (s_load).
// Data is touched exactly once (201 MB each way > reuse horizon), so use
// non-temporal hints on the bulk stream.
// ---------------------------------------------------------------------------
__global__ void cca_apply(const v4f* __restrict__ x,
                          const float* __restrict__ scale,
                          v4f* __restrict__ out) {
    const int i = blockIdx.x * 256 + threadIdx.x;
    // 4096 floats (=1024 float4) per (b,c) slab; a 256-thread block never
    // crosses a 1024-float4 boundary fractionally: channel uniform per block.
    const int c = (blockIdx.x >> 2) % C_CHANNELS;
    const float s = scale[c];

    v4f v = __builtin_nontemporal_load(&x[i]);
    v.x *= s; v.y *= s; v.z *= s; v.w *= s;
    __builtin_nontemporal_store(v, &out[i]);
}

// ---------------------------------------------------------------------------
extern "C" void kernel_launch(void* const* d_in, const int* in_sizes, int n_in,
                              void* d_out, int out_size, void* d_ws, size_t ws_size,
                              hipStream_t stream) {
    const float* x    = (const float*)d_in[0];   // (16, 768, 64, 64)
    const float* attn = (const float*)d_in[1];   // (150, 768)
    float* out = (float*)d_out;

    float* m     = (float*)d_ws;                 // 150 floats
    float* invd  = m + K_CLASSES;                // 150 floats
    float* scale = invd + K_CLASSES;             // 768 floats

    cca_rowstats<<<K_CLASSES, 32, 0, stream>>>(attn, m, invd);
    cca_scale_wmma<<<C_CHANNELS / 16, 32, 0, stream>>>(attn, m, invd, scale);

    const int n4 = out_size / 4;                 // 12,582,912 float4
    cca_apply<<<n4 / 256, 256, 0, stream>>>((const v4f*)x, scale, (v4f*)out);
}